// NonLocal_80513456930969
// MI455X (gfx1250) — compile-verified
//
#include <hip/hip_runtime.h>
#include <math.h>

typedef __attribute__((ext_vector_type(2))) float v2f;
typedef __attribute__((ext_vector_type(8))) float v8f;

#define BM 128
#define BN 128
#define KC 16
#define KPA 17    // padded A row (k contiguous); 17 coprime with 64 banks
#define BNP 136   // padded B row (n contiguous); 544B row: 16B-aligned, half-wave banks disjoint

__device__ __forceinline__ void async_copy_b128(unsigned lds_byte_addr, const float* gaddr) {
    // GLOBAL_LOAD_ASYNC_TO_LDS_B128: per-lane 16B global -> LDS, tracked by ASYNCcnt
    asm volatile("global_load_async_to_lds_b128 %0, %1, off"
                 :: "v"(lds_byte_addr), "v"(gaddr)
                 : "memory");
}
__device__ __forceinline__ void wait_async0() {
    asm volatile("s_wait_asynccnt 0x0" ::: "memory");
}

// Generic fp32 WMMA GEMM with double-buffered LDS + async B-tile copies:
//   C[b][m][n] = sum_k A[b][m][k]*B[b][k][n] (+ R[b][m][n])
// A(m,k) at A + b*a_sb + m*a_sm + k*a_sk   (either a_sk==1 or a_sm==1)
// B(k,n) at B + b*b_sb + k*b_sk + n        (n always contiguous)
// C(m,n) at C + b*c_sb + m*c_sm + n        (n always contiguous); R uses C strides.
// M%128==0, N%128==0, K%16==0, b_sk%4==0.
__global__ __launch_bounds__(256)
void gemm_f32_wmma(const float* __restrict__ A, int a_sm, int a_sk, long long a_sb,
                   const float* __restrict__ B, int b_sk, long long b_sb,
                   float* __restrict__ C, int c_sm, long long c_sb,
                   const float* __restrict__ R, long long r_sb,
                   int M, int N, int K)
{
    __shared__ float As[2][BM * KPA];   // As[m][k]
    __shared__ float Bs[2][KC * BNP];   // Bs[k][n]

    const int bz = blockIdx.z;
    const float* Ab = A + (long long)bz * a_sb;
    const float* Bb = B + (long long)bz * b_sb;
    float*       Cb = C + (long long)bz * c_sb;

    const int m0   = blockIdx.y * BM;
    const int n0   = blockIdx.x * BN;
    const int tid  = threadIdx.x;
    const int lane = tid & 31;
    const int wave = tid >> 5;
    const int wm   = (wave & 3) * 32;   // 4 waves down, 32 rows each
    const int wn   = (wave >> 2) * 64;  // 2 waves across, 64 cols each
    const int half = lane >> 4;
    const int lm   = lane & 15;

    // ---- per-thread A-tile element indices (8 per thread), 32-bit offsets ----
    int am[8], ak[8];
    if (a_sk == 1) {            // A rows k-contiguous: k fastest
        #pragma unroll
        for (int i = 0; i < 8; ++i) { int idx = tid + i * 256; ak[i] = idx & 15;  am[i] = idx >> 4; }
    } else {                    // a_sm==1: m fastest for coalescing
        #pragma unroll
        for (int i = 0; i < 8; ++i) { int idx = tid + i * 256; am[i] = idx & 127; ak[i] = idx >> 7; }
    }
    int aoff[8];
    #pragma unroll
    for (int i = 0; i < 8; ++i) aoff[i] = (m0 + am[i]) * a_sm + ak[i] * a_sk;

    // ---- per-thread B async-copy slots (2 x b128 per thread per chunk) ----
    int bk[2], bn4[2];
    #pragma unroll
    for (int j = 0; j < 2; ++j) { int idx = tid + j * 256; bk[j] = idx >> 5; bn4[j] = (idx & 31) * 4; }

    v8f acc[2][4] = {};
    float areg[8];

    // ---- prologue: stage chunk 0 into buffer 0 ----
    #pragma unroll
    for (int j = 0; j < 2; ++j) {
        const float* g = Bb + bk[j] * b_sk + (n0 + bn4[j]);
        async_copy_b128((unsigned)(unsigned long long)&Bs[0][bk[j] * BNP + bn4[j]], g);
    }
    #pragma unroll
    for (int i = 0; i < 8; ++i) areg[i] = Ab[aoff[i]];
    #pragma unroll
    for (int i = 0; i < 8; ++i) As[0][am[i] * KPA + ak[i]] = areg[i];
    wait_async0();
    __syncthreads();

    int buf = 0;
    for (int k0 = 0; k0 < K; k0 += KC) {
        const int nxt = buf ^ 1;
        const bool more = (k0 + KC) < K;

        // ---- prefetch next chunk: async B -> Bs[nxt], A -> regs ----
        if (more) {
            #pragma unroll
            for (int j = 0; j < 2; ++j) {
                const float* g = Bb + (k0 + KC + bk[j]) * b_sk + (n0 + bn4[j]);
                async_copy_b128((unsigned)(unsigned long long)&Bs[nxt][bk[j] * BNP + bn4[j]], g);
            }
            #pragma unroll
            for (int i = 0; i < 8; ++i) areg[i] = Ab[aoff[i] + (k0 + KC) * a_sk];
        }

        // ---- compute current chunk: 32 WMMA per wave ----
        const float* a_base = &As[buf][0];
        const float* b_base = &Bs[buf][0];
        #pragma unroll
        for (int kk = 0; kk < KC; kk += 4) {
            v2f af[2], bf[4];
            #pragma unroll
            for (int tm = 0; tm < 2; ++tm) {
                const float* p = a_base + (wm + tm * 16 + lm) * KPA + kk + half * 2;
                af[tm] = v2f{p[0], p[1]};
            }
            #pragma unroll
            for (int tn = 0; tn < 4; ++tn) {
                const float* p = b_base + (kk + half * 2) * BNP + wn + tn * 16 + lm;
                bf[tn] = v2f{p[0], p[BNP]};
            }
            #pragma unroll
            for (int tm = 0; tm < 2; ++tm)
                #pragma unroll
                for (int tn = 0; tn < 4; ++tn)
                    acc[tm][tn] = __builtin_amdgcn_wmma_f32_16x16x4_f32(
                        false, af[tm], false, bf[tn], (short)0, acc[tm][tn], false, false);
        }

        if (more) {
            #pragma unroll
            for (int i = 0; i < 8; ++i) As[nxt][am[i] * KPA + ak[i]] = areg[i];
        }
        wait_async0();
        __syncthreads();
        buf = nxt;
    }

    // ---- epilogue: C 16x16 layout (vgpr i -> row half*8+i, col lm) ----
    const float* Rb = R ? (R + (long long)bz * r_sb) : nullptr;
    #pragma unroll
    for (int tm = 0; tm < 2; ++tm)
        #pragma unroll
        for (int tn = 0; tn < 4; ++tn)
            #pragma unroll
            for (int i = 0; i < 8; ++i) {
                int m = m0 + wm + tm * 16 + half * 8 + i;
                int n = n0 + wn + tn * 16 + lm;
                int off = m * c_sm + n;
                float v = acc[tm][tn][i];
                if (Rb) v += Rb[off];
                Cb[off] = v;
            }
}

// In-place row softmax: one 256-thread block per contiguous row of `ncols` f32.
__global__ __launch_bounds__(256)
void softmax_rows(float* __restrict__ S, int ncols)
{
    __shared__ float red[256];
    float* row = S + (long long)blockIdx.x * ncols;
    const int tid = threadIdx.x;

    float mx = -INFINITY;
    for (int i = tid; i < ncols; i += 256) mx = fmaxf(mx, row[i]);
    red[tid] = mx; __syncthreads();
    for (int s = 128; s > 0; s >>= 1) {
        if (tid < s) red[tid] = fmaxf(red[tid], red[tid + s]);
        __syncthreads();
    }
    mx = red[0]; __syncthreads();

    float sum = 0.f;
    for (int i = tid; i < ncols; i += 256) {
        float e = __expf(row[i] - mx);
        row[i] = e;
        sum += e;
    }
    red[tid] = sum; __syncthreads();
    for (int s = 128; s > 0; s >>= 1) {
        if (tid < s) red[tid] += red[tid + s];
        __syncthreads();
    }
    const float inv = 1.0f / red[0];
    for (int i = tid; i < ncols; i += 256) row[i] *= inv;
}

extern "C" void kernel_launch(void* const* d_in, const int* in_sizes, int n_in,
                              void* d_out, int out_size, void* d_ws, size_t ws_size,
                              hipStream_t stream)
{
    const float* x       = (const float*)d_in[0]; // (8,512,64,64)
    const float* w_phi   = (const float*)d_in[1]; // (256,512)
    const float* w_theta = (const float*)d_in[2];
    const float* w_g     = (const float*)d_in[3];
    const float* w_mask  = (const float*)d_in[4]; // (512,256)
    float* out = (float*)d_out;                   // (8,512,64,64)
    float* ws  = (float*)d_ws;

    const long long PB = 1048576;   // per-batch conv-out elements (256*4096)==(512*2048)
    const long long XB = 2097152;   // per-batch x elements (512*4096)
    const long long SB = 4194304;   // per-batch score elements (2048*2048)

    float* phiR   = ws;               // 8 * PB
    float* thetaR = ws + 8  * PB;     // 8 * PB
    float* gR     = ws + 16 * PB;     // 8 * PB
    float* S      = ws + 24 * PB;     // 8 * SB
    float* Tb     = phiR;             // reuse: phi dead after score GEMM

    dim3 blk(256);

    // 1) qkv convs: M=256, N=4096, K=512. A = W (row-major), B = x_b (512x4096).
    {
        dim3 grd(4096 / BN, 256 / BM, 8);
        gemm_f32_wmma<<<grd, blk, 0, stream>>>(w_phi,   512, 1, 0,  x, 4096, XB,
                                               phiR,   4096, PB, nullptr, 0, 256, 4096, 512);
        gemm_f32_wmma<<<grd, blk, 0, stream>>>(w_theta, 512, 1, 0,  x, 4096, XB,
                                               thetaR, 4096, PB, nullptr, 0, 256, 4096, 512);
        gemm_f32_wmma<<<grd, blk, 0, stream>>>(w_g,     512, 1, 0,  x, 4096, XB,
                                               gR,     4096, PB, nullptr, 0, 256, 4096, 512);
    }

    // 2) scores (stored transposed): S[m][n] = sum_c phi[c*2048+m]*theta[c*2048+n] == attn[n][m]
    //    -> torch softmax(dim=1) becomes a contiguous row softmax of S.
    {
        dim3 grd(2048 / BN, 2048 / BM, 8);
        gemm_f32_wmma<<<grd, blk, 0, stream>>>(phiR, 1, 2048, PB,  thetaR, 2048, PB,
                                               S, 2048, SB, nullptr, 0, 2048, 2048, 512);
    }

    // 3) softmax over rows of S (8*2048 rows of 2048), in place
    softmax_rows<<<dim3(8 * 2048), blk, 0, stream>>>(S, 2048);

    // 4) apply + transpose in one GEMM: Tb[c][n] = sum_m g_flat[c*2048+m]*S[m][n]
    //    Tb (512,2048) contiguous == the (256,4096) view the mask conv needs.
    {
        dim3 grd(2048 / BN, 512 / BM, 8);
        gemm_f32_wmma<<<grd, blk, 0, stream>>>(gR, 2048, 1, PB,  S, 2048, SB,
                                               Tb, 2048, PB, nullptr, 0, 512, 2048, 2048);
    }

    // 5) mask conv + residual: out[b][oc][p] = sum_ic w_mask[oc][ic]*Tb_b[ic*4096+p] + x
    {
        dim3 grd(4096 / BN, 512 / BM, 8);
        gemm_f32_wmma<<<grd, blk, 0, stream>>>(w_mask, 256, 1, 0,  Tb, 4096, PB,
                                               out, 4096, XB, x, XB, 512, 4096, 256);
    }
}